// SparseNeuralNetworkArchitectureX_27573690040595
// MI455X (gfx1250) — compile-verified
//
#include <hip/hip_runtime.h>
#include <cstddef>

#define IN_F  768
#define HID   64
#define BATCH 131072

#define WAVES       8
#define ROWS_PER_WG (WAVES * 16)
#define HTS         68    // htile row stride in floats (bank-spread padding)
#define KPS         136   // LDS stride per k-pair (floats): 16 lanes*8 + pad

typedef float v2f __attribute__((ext_vector_type(2)));
typedef float v4f __attribute__((ext_vector_type(4)));
typedef float v8f __attribute__((ext_vector_type(8)));

// LDS layout offsets (in floats)
#define W1I_OFF 0
#define W1I_SZ  ((IN_F / 2) * KPS)            // 52224
#define W2I_OFF (W1I_OFF + W1I_SZ)            // 52224
#define W2I_SZ  ((HID / 2) * KPS)             // 4352
#define W3S_OFF (W2I_OFF + W2I_SZ)            // 56576
#define B1S_OFF (W3S_OFF + 64)                // 56640
#define B2S_OFF (B1S_OFF + 64)                // 56704
#define HT_OFF  (B2S_OFF + 64)                // 56768
#define SMEM_FLOATS (HT_OFF + WAVES * 16 * HTS)

// ---------------------------------------------------------------------------
// Densify COO weights into d_ws (duplicates accumulate, like torch coalesce).
// ---------------------------------------------------------------------------
__global__ void zero_ws_kernel(float* __restrict__ w, int n) {
    for (int i = blockIdx.x * blockDim.x + threadIdx.x; i < n;
         i += gridDim.x * blockDim.x)
        w[i] = 0.0f;
}

__global__ void scatter_coo_kernel(const int* __restrict__ idx1, const float* __restrict__ val1, int nnz1,
                                   const int* __restrict__ idx2, const float* __restrict__ val2, int nnz2,
                                   const int* __restrict__ idx3, const float* __restrict__ val3, int nnz3,
                                   float* __restrict__ W1d, float* __restrict__ W2d, float* __restrict__ W3d) {
    int i = blockIdx.x * blockDim.x + threadIdx.x;
    if (i < nnz1) {
        atomicAdd(&W1d[idx1[i] * HID + idx1[nnz1 + i]], val1[i]);
    } else if (i < nnz1 + nnz2) {
        int j = i - nnz1;
        atomicAdd(&W2d[idx2[j] * HID + idx2[nnz2 + j]], val2[j]);
    } else if (i < nnz1 + nnz2 + nnz3) {
        int j = i - nnz1 - nnz2;
        // OUT == 1, so column index is always 0; r*1 + c:
        atomicAdd(&W3d[idx3[j] + idx3[nnz3 + j]], val3[j]);
    }
}

// Interleaved LDS address for weight element (row k, col c):
//   pairs both k-parities next to each other so one ds_load_b128 yields two
//   complete B fragments with no register shuffles.
__device__ __forceinline__ int wperm(int k, int c) {
    int lo = c & 15;
    return (k >> 1) * KPS + lo * 8 + ((lo >> 3) << 2) + ((c >> 4) << 1) + (k & 1);
}

// ---------------------------------------------------------------------------
// Fused 3-layer MLP. One wave = 16 rows of x. W1/W2 dense in LDS.
// ---------------------------------------------------------------------------
__global__ __launch_bounds__(256) void fused_mlp_kernel(
        const float* __restrict__ x,
        const float* __restrict__ W1d, const float* __restrict__ W2d,
        const float* __restrict__ W3d,
        const float* __restrict__ b1, const float* __restrict__ b2,
        const float* __restrict__ b3,
        float* __restrict__ out) {
    extern __shared__ float smem[];
    float* W1i = smem + W1I_OFF;
    float* W2i = smem + W2I_OFF;
    float* W3s = smem + W3S_OFF;
    float* b1s = smem + B1S_OFF;
    float* b2s = smem + B2S_OFF;
    float* ht  = smem + HT_OFF;

    const int tid = threadIdx.x;

    // Coalesced global reads, interleaved LDS writes.
    for (int i = tid; i < IN_F * HID; i += 256)
        W1i[wperm(i >> 6, i & 63)] = W1d[i];
    for (int i = tid; i < HID * HID; i += 256)
        W2i[wperm(i >> 6, i & 63)] = W2d[i];
    if (tid < 64) {
        W3s[tid] = W3d[tid];
        b1s[tid] = b1[tid];
        b2s[tid] = b2[tid];
    }
    __syncthreads();

    const int wave = tid >> 5;
    const int lane = tid & 31;
    const int lo   = lane & 15;   // M index (A) or N index (B/C)
    const int hi   = lane >> 4;   // k-half selector
    const int row0 = blockIdx.x * ROWS_PER_WG + wave * 16;

    const float* xrow = x + (size_t)(row0 + lo) * IN_F + 2 * hi;
    float* hw = ht + wave * (16 * HTS);

    // Per-lane base into the interleaved weight layout.
    const int laneoff = lo * 8 + ((lo >> 3) << 2) + hi * KPS;

    // ---------------- Layer 1: [16 x 768] @ [768 x 64] -----------------
    v8f acc[4] = {};
#pragma unroll 4
    for (int k0 = 0; k0 < IN_F; k0 += 4) {
        // A fragment: lane<16 -> x[m][k0..k0+1]; lane>=16 -> x[m][k0+2..k0+3]
        v2f a = *(const v2f*)(xrow + k0);
        const int boff = (k0 >> 1) * KPS + laneoff;
        v4f q0 = *(const v4f*)&W1i[boff];       // tiles 0,1 (both k parities)
        v4f q1 = *(const v4f*)&W1i[boff + 4];   // tiles 2,3
        acc[0] = __builtin_amdgcn_wmma_f32_16x16x4_f32(false, a, false,
                     __builtin_shufflevector(q0, q0, 0, 1), (short)0, acc[0], false, false);
        acc[1] = __builtin_amdgcn_wmma_f32_16x16x4_f32(false, a, false,
                     __builtin_shufflevector(q0, q0, 2, 3), (short)0, acc[1], false, false);
        acc[2] = __builtin_amdgcn_wmma_f32_16x16x4_f32(false, a, false,
                     __builtin_shufflevector(q1, q1, 0, 1), (short)0, acc[2], false, false);
        acc[3] = __builtin_amdgcn_wmma_f32_16x16x4_f32(false, a, false,
                     __builtin_shufflevector(q1, q1, 2, 3), (short)0, acc[3], false, false);
    }

    // bias + relu, write h1 tile to LDS (C layout: m = r + 8*hi, n = 16t + lo)
#pragma unroll
    for (int t = 0; t < 4; ++t) {
        const float bb = b1s[16 * t + lo];
#pragma unroll
        for (int r = 0; r < 8; ++r) {
            float v = acc[t][r] + bb;
            v = v > 0.0f ? v : 0.0f;
            hw[(r + 8 * hi) * HTS + 16 * t + lo] = v;
        }
    }

    // ---------------- Layer 2: [16 x 64] @ [64 x 64] -------------------
    v8f acc2[4] = {};
#pragma unroll 2
    for (int k0 = 0; k0 < HID; k0 += 4) {
        const int kb = k0 + 2 * hi;
        v2f a = *(const v2f*)&hw[lo * HTS + kb];
        const int boff = (k0 >> 1) * KPS + laneoff;
        v4f q0 = *(const v4f*)&W2i[boff];
        v4f q1 = *(const v4f*)&W2i[boff + 4];
        acc2[0] = __builtin_amdgcn_wmma_f32_16x16x4_f32(false, a, false,
                      __builtin_shufflevector(q0, q0, 0, 1), (short)0, acc2[0], false, false);
        acc2[1] = __builtin_amdgcn_wmma_f32_16x16x4_f32(false, a, false,
                      __builtin_shufflevector(q0, q0, 2, 3), (short)0, acc2[1], false, false);
        acc2[2] = __builtin_amdgcn_wmma_f32_16x16x4_f32(false, a, false,
                      __builtin_shufflevector(q1, q1, 0, 1), (short)0, acc2[2], false, false);
        acc2[3] = __builtin_amdgcn_wmma_f32_16x16x4_f32(false, a, false,
                      __builtin_shufflevector(q1, q1, 2, 3), (short)0, acc2[3], false, false);
    }

    // bias + relu, write h2 back into the same per-wave LDS tile
    // (same-wave LDS ops complete in order: all h1 reads precede these writes)
#pragma unroll
    for (int t = 0; t < 4; ++t) {
        const float bb = b2s[16 * t + lo];
#pragma unroll
        for (int r = 0; r < 8; ++r) {
            float v = acc2[t][r] + bb;
            v = v > 0.0f ? v : 0.0f;
            hw[(r + 8 * hi) * HTS + 16 * t + lo] = v;
        }
    }

    // ---------------- Layer 3: [16 x 64] @ [64 x 1] --------------------
    float part = 0.0f;
    const int nbase = 32 * hi;
#pragma unroll 8
    for (int j = 0; j < 32; ++j)
        part += hw[lo * HTS + nbase + j] * W3s[nbase + j];
    part += __shfl_xor(part, 16);
    if (hi == 0)
        out[row0 + lo] = part + b3[0];
}

// ---------------------------------------------------------------------------
extern "C" void kernel_launch(void* const* d_in, const int* in_sizes, int n_in,
                              void* d_out, int out_size, void* d_ws, size_t ws_size,
                              hipStream_t stream) {
    const float* x    = (const float*)d_in[0];
    const int*   idx1 = (const int*)  d_in[1];
    const float* val1 = (const float*)d_in[2];
    const float* b1   = (const float*)d_in[3];
    const int*   idx2 = (const int*)  d_in[4];
    const float* val2 = (const float*)d_in[5];
    const float* b2   = (const float*)d_in[6];
    const int*   idx3 = (const int*)  d_in[7];
    const float* val3 = (const float*)d_in[8];
    const float* b3   = (const float*)d_in[9];

    const int nnz1 = in_sizes[2];
    const int nnz2 = in_sizes[5];
    const int nnz3 = in_sizes[8];

    float* W1d = (float*)d_ws;            // 768*64
    float* W2d = W1d + IN_F * HID;        // 64*64
    float* W3d = W2d + HID * HID;         // 64

    const int ntot = IN_F * HID + HID * HID + HID;
    zero_ws_kernel<<<(ntot + 255) / 256, 256, 0, stream>>>(W1d, ntot);

    const int ns = nnz1 + nnz2 + nnz3;
    scatter_coo_kernel<<<(ns + 255) / 256, 256, 0, stream>>>(
        idx1, val1, nnz1, idx2, val2, nnz2, idx3, val3, nnz3, W1d, W2d, W3d);

    const size_t shmem = (size_t)SMEM_FLOATS * sizeof(float); // ~262 KB < 320 KB
    fused_mlp_kernel<<<BATCH / ROWS_PER_WG, 256, shmem, stream>>>(
        x, W1d, W2d, W3d, b1, b2, b3, (float*)d_out);
}